// WideBNetModel_13048110646038
// MI455X (gfx1250) — compile-verified
//
#include <hip/hip_runtime.h>
#include <hip/hip_bf16.h>

// WideBNet on MI455X (gfx1250).
// Memory-bound weight streaming (~500MB fp32 weights @ 23.3 TB/s ~= 21us,
// FLOPs ~1.7G are negligible). Core primitive: complex position-dependent
// batched GEMM done with V_WMMA_F32_16X16X4_F32, batch(8) in rows 0-7,
// 16 output channels per wave, two f32 accumulators (real/imag).
// Weight loads are non-temporal (streamed once); activations stay in L2.

typedef float v2f __attribute__((ext_vector_type(2)));
typedef float v8f __attribute__((ext_vector_type(8)));

#define CDIV(a,b) (((a)+(b)-1)/(b))

// ---------------------------------------------------------------------------
// Complex block GEMM:
//   x:  (16, G, K)  rows (b*2+ri) activations, contiguous K
//   w:  (4, G, K, O) four real matrices w0..w3 (complex block)
//   y:  (16, G, O)
//   yr = xr*w0 + xi*w1 ; yi = xr*w2 + xi*w3
// Epilogue: optional resid (y = 2*resid + acc), optional relu.
// One wave per (o-tile, g). Weight dwords read exactly once, coalesced, NT.
// No divergence in the K loop: out-of-range rows/cols are clamped to valid
// addresses; the resulting duplicate accumulator rows/cols are never stored.
// ---------------------------------------------------------------------------
__global__ __launch_bounds__(32)
void cblock_wmma(const float* __restrict__ x, const float* __restrict__ w,
                 float* __restrict__ y, const float* __restrict__ resid,
                 int G, int K, int O, int addResid, int doRelu)
{
    const int g    = blockIdx.y;
    const int o0   = blockIdx.x * 16;
    const int lane = threadIdx.x;   // 0..31
    const int row  = lane & 15;     // A: M row (batch).  B: N col (out chan)
    const int half = lane >> 4;     // selects K sub-pair {0,1} vs {2,3}
    const int o    = o0 + row;
    const bool oval = (o < O);
    const int oc   = oval ? o : (O - 1);  // clamped col (dupe data, never stored)
    const int rc   = row & 7;             // clamped batch row (rows 8-15 dupe 0-7)

    const size_t qs   = (size_t)G * K * O;       // stride between w0..w3
    const size_t wrow = (size_t)4 * O;           // K-chunk advance in w
    const float* pw0 = w + (size_t)g * K * O + (size_t)(half * 2) * O + oc;
    const float* pw1 = pw0 + qs;
    const float* pw2 = pw1 + qs;
    const float* pw3 = pw2 + qs;
    const float* pxr = x + ((size_t)(rc * 2 + 0) * G + g) * K + half * 2;
    const float* pxi = x + ((size_t)(rc * 2 + 1) * G + g) * K + half * 2;

    v8f accR = {};
    v8f accI = {};

#pragma unroll 2
    for (int kc = 0; kc < K; kc += 4) {
        const v2f ar = *(const v2f*)pxr;
        const v2f ai = *(const v2f*)pxi;
        const v2f b0 = { __builtin_nontemporal_load(pw0), __builtin_nontemporal_load(pw0 + O) };
        const v2f b1 = { __builtin_nontemporal_load(pw1), __builtin_nontemporal_load(pw1 + O) };
        const v2f b2 = { __builtin_nontemporal_load(pw2), __builtin_nontemporal_load(pw2 + O) };
        const v2f b3 = { __builtin_nontemporal_load(pw3), __builtin_nontemporal_load(pw3 + O) };
        // D = A x B + C  (fp32, 16x16x4)
        accR = __builtin_amdgcn_wmma_f32_16x16x4_f32(false, ar, false, b0, (short)0, accR, false, false);
        accR = __builtin_amdgcn_wmma_f32_16x16x4_f32(false, ai, false, b1, (short)0, accR, false, false);
        accI = __builtin_amdgcn_wmma_f32_16x16x4_f32(false, ar, false, b2, (short)0, accI, false, false);
        accI = __builtin_amdgcn_wmma_f32_16x16x4_f32(false, ai, false, b3, (short)0, accI, false, false);
        pxr += 4; pxi += 4;
        pw0 += wrow; pw1 += wrow; pw2 += wrow; pw3 += wrow;
    }

    // C/D layout: element (M = v + 8*(lane>=16), N = lane&15).
    // Lanes 0-15 hold the valid batch rows 0-7 in vgprs v=0..7.
    if (lane < 16 && oval) {
#pragma unroll
        for (int b = 0; b < 8; ++b) {
            float vr = accR[b], vi = accI[b];
            const size_t ir = ((size_t)(b * 2 + 0) * G + g) * O + o;
            const size_t ii = ((size_t)(b * 2 + 1) * G + g) * O + o;
            if (addResid) { vr = 2.f * resid[ir] + vr; vi = 2.f * resid[ii] + vi; }
            if (doRelu)   { vr = fmaxf(vr, 0.f);       vi = fmaxf(vi, 0.f); }
            y[ir] = vr;
            y[ii] = vi;
        }
    }
}

// ---------------------------------------------------------------------------
// Gather / reshape kernels (all index math derived in closed form)
// perm(l=3) == identity; perm(l=2)[n] = (n&~15) | ((n%4)<<2) | ((n%16)>>2)
// switch[n] = (n%16)*16 + n/16
// ---------------------------------------------------------------------------
__device__ __forceinline__ int perm2(int n) {
    const int rem = n & 15;
    return (n & ~15) | ((rem & 3) << 2) | (rem >> 2);
}

// x_d[i] = inputs[3*i + d]   (slices inputs[...,d]; reshape is flat)
__global__ void strided_pick(const float* __restrict__ in, float* __restrict__ out,
                             int n, int stride, int off)
{
    const int i = blockIdx.x * blockDim.x + threadIdx.x;
    if (i < n) out[i] = in[(size_t)i * stride + off];
}

// xH0 (16,64,192): concat(y0 (16,256,24), repeat4(y1 (16,64,24))) then k=4 fold
__global__ void build_xH0(const float* __restrict__ y0, const float* __restrict__ y1,
                          float* __restrict__ out)
{
    const int i = blockIdx.x * blockDim.x + threadIdx.x;
    if (i >= 16 * 64 * 192) return;
    const int kk = i % 192;
    const int g  = (i / 192) % 64;
    const int br = i / (192 * 64);
    const int n  = g * 4 + kk / 48;     // perm(l=3) = identity
    const int c  = kk % 48;
    out[i] = (c < 24) ? y0[((size_t)br * 256 + n) * 24 + c]
                      : y1[((size_t)br * 64 + n / 4) * 24 + (c - 24)];
}

// xH1 (16,64,288): concat(yH0-view (16,256,48), repeat16(y2 (16,16,24))),
// permuted by perm2, then k=4 fold
__global__ void build_xH1(const float* __restrict__ yH0, const float* __restrict__ y2,
                          float* __restrict__ out)
{
    const int i = blockIdx.x * blockDim.x + threadIdx.x;
    if (i >= 16 * 64 * 288) return;
    const int kk = i % 288;
    const int g  = (i / 288) % 64;
    const int br = i / (288 * 64);
    const int n  = g * 4 + kk / 72;
    const int c  = kk % 72;
    const int np = perm2(n);
    out[i] = (c < 48) ? yH0[((size_t)br * 64 + np / 4) * 192 + (np % 4) * 48 + c]
                      : y2[((size_t)br * 16 + np / 16) * 24 + (c - 48)];
}

// ysw (16,256,72) = switch-take of yH1 (16,64,288) viewed as (16,256,72)
__global__ void build_ysw(const float* __restrict__ yH1, float* __restrict__ out)
{
    const int i = blockIdx.x * blockDim.x + threadIdx.x;
    if (i >= 16 * 256 * 72) return;
    const int c  = i % 72;
    const int n  = (i / 72) % 256;
    const int br = i / (72 * 256);
    const int sw = (n % 16) * 16 + n / 16;
    out[i] = yH1[((size_t)br * 64 + sw / 4) * 288 + (sw % 4) * 72 + c];
}

// xG1 (16,64,288): perm2-take of yG0 (16,64,288)-as-(16,256,72), then k=4 fold
__global__ void build_xG1(const float* __restrict__ yG0, float* __restrict__ out)
{
    const int i = blockIdx.x * blockDim.x + threadIdx.x;
    if (i >= 16 * 64 * 288) return;
    const int kk = i % 288;
    const int g  = (i / 288) % 64;
    const int br = i / (288 * 64);
    const int n  = g * 4 + kk / 72;
    const int c  = kk % 72;
    const int np = perm2(n);
    out[i] = yG0[((size_t)br * 64 + np / 4) * 288 + (np % 4) * 72 + c];
}

// img0 (8,128,128) = morton gather of real part of yU (16,256,64)
__global__ void morton_gather(const float* __restrict__ yU, float* __restrict__ out)
{
    const int i = blockIdx.x * blockDim.x + threadIdx.x;
    if (i >= 8 * 128 * 128) return;
    const int j  = i % 128;
    const int ii = (i / 128) % 128;
    const int b  = i / (128 * 128);
    const int pi = ii >> 3, pj = j >> 3;
    int pz = 0;
#pragma unroll
    for (int k = 0; k < 4; ++k) {
        pz |= ((pi >> k) & 1) << (2 * k + 1);
        pz |= ((pj >> k) & 1) << (2 * k);
    }
    const int pos = pz * 64 + (ii & 7) * 8 + (j & 7);
    out[i] = yU[(size_t)(b * 2) * 16384 + pos];
}

// 2x2 'SAME' conv (jax pads 0 before, 1 after), NHWC, small channel counts
__global__ void conv2x2(const float* __restrict__ in, const float* __restrict__ w,
                        const float* __restrict__ bias, float* __restrict__ out,
                        int Cin, int Cout, int doRelu)
{
    const int i = blockIdx.x * blockDim.x + threadIdx.x;
    if (i >= 8 * 128 * 128 * Cout) return;
    const int co = i % Cout;
    const int j  = (i / Cout) % 128;
    const int ii = (i / (Cout * 128)) % 128;
    const int b  = i / (Cout * 128 * 128);
    float acc = bias[co];
    for (int di = 0; di < 2; ++di) {
        const int yy = ii + di;
        if (yy >= 128) continue;
        for (int dj = 0; dj < 2; ++dj) {
            const int xx = j + dj;
            if (xx >= 128) continue;
            for (int ci = 0; ci < Cin; ++ci)
                acc += in[(((size_t)b * 128 + yy) * 128 + xx) * Cin + ci] *
                       w[((di * 2 + dj) * Cin + ci) * Cout + co];
        }
    }
    if (doRelu) acc = fmaxf(acc, 0.f);
    out[i] = acc;
}

// ---------------------------------------------------------------------------
extern "C" void kernel_launch(void* const* d_in, const int* in_sizes, int n_in,
                              void* d_out, int out_size, void* d_ws, size_t ws_size,
                              hipStream_t stream)
{
    (void)in_sizes; (void)n_in; (void)out_size; (void)ws_size;

    const float* inputs = (const float*)d_in[0];
    const float* V0w = (const float*)d_in[1];   // (4,256,64,24)
    const float* V1w = (const float*)d_in[2];   // (4,64,256,24)
    const float* V2w = (const float*)d_in[3];   // (4,16,1024,24)
    const float* H0w = (const float*)d_in[4];   // (4,64,192,192)
    const float* H1w = (const float*)d_in[5];   // (4,64,288,288)
    const float* Mw  = (const float*)d_in[6];   // (3,2,4,256,72,72)
    const float* G0w = (const float*)d_in[7];   // (4,64,288,288)
    const float* G1w = (const float*)d_in[8];   // (4,64,288,192)
    const float* Uw  = (const float*)d_in[9];   // (4,256,48,64)
    const float* cw0 = (const float*)d_in[10];
    const float* cw1 = (const float*)d_in[11];
    const float* cb0 = (const float*)d_in[12];
    const float* cb1 = (const float*)d_in[13];
    const float* fw  = (const float*)d_in[14];
    const float* fb  = (const float*)d_in[15];
    float* out = (float*)d_out;

    // workspace carve-up (floats); total ~5.54M floats (~22.1 MB)
    float* ws = (float*)d_ws;
    size_t off = 0;
    auto alloc = [&](size_t n) { float* p = ws + off; off += n; return p; };
    float* x0   = alloc(262144);  // (16,256,64)
    float* y0   = alloc(98304);   // (16,256,24)
    float* x1   = alloc(262144);  // (16,64,256)
    float* y1   = alloc(24576);   // (16,64,24)
    float* x2   = alloc(262144);  // (16,16,1024)
    float* y2   = alloc(6144);    // (16,16,24)
    float* xH0  = alloc(196608);  // (16,64,192)
    float* yH0  = alloc(196608);  // (16,64,192)
    float* xH1  = alloc(294912);  // (16,64,288)
    float* yH1  = alloc(294912);  // (16,64,288)
    float* bufA = alloc(294912);  // (16,256,72) resnet ping
    float* bufB = alloc(294912);  // (16,256,72) resnet pong
    float* tmpT = alloc(294912);  // (16,256,72) resnet inner
    float* yG0  = alloc(294912);  // (16,64,288)
    float* xG1  = alloc(294912);  // (16,64,288)
    float* yG1  = alloc(196608);  // (16,64,192) == (16,256,48) contiguous
    float* yU   = alloc(262144);  // (16,256,64)
    float* img0 = alloc(131072);  // (8,128,128,1)
    float* img1 = alloc(786432);  // (8,128,128,6)
    float* img2 = alloc(786432);  // (8,128,128,6)

    const dim3 blk(256);
    auto grid1 = [](int n) { return dim3((unsigned)CDIV(n, 256)); };

    // ---- V stages (inputs[...,d] slices are stride-3 picks) ----
    strided_pick<<<grid1(262144), blk, 0, stream>>>(inputs, x0, 262144, 3, 0);
    cblock_wmma<<<dim3(2, 256), dim3(32), 0, stream>>>(x0, V0w, y0, nullptr, 256, 64, 24, 0, 0);
    strided_pick<<<grid1(262144), blk, 0, stream>>>(inputs, x1, 262144, 3, 1);
    cblock_wmma<<<dim3(2, 64), dim3(32), 0, stream>>>(x1, V1w, y1, nullptr, 64, 256, 24, 0, 0);
    strided_pick<<<grid1(262144), blk, 0, stream>>>(inputs, x2, 262144, 3, 2);
    cblock_wmma<<<dim3(2, 16), dim3(32), 0, stream>>>(x2, V2w, y2, nullptr, 16, 1024, 24, 0, 0);

    // ---- H stages ----
    build_xH0<<<grid1(196608), blk, 0, stream>>>(y0, y1, xH0);
    cblock_wmma<<<dim3(12, 64), dim3(32), 0, stream>>>(xH0, H0w, yH0, nullptr, 64, 192, 192, 0, 0);
    build_xH1<<<grid1(294912), blk, 0, stream>>>(yH0, y2, xH1);
    cblock_wmma<<<dim3(18, 64), dim3(32), 0, stream>>>(xH1, H1w, yH1, nullptr, 64, 288, 288, 0, 0);

    // ---- switch + resnet (y_new = 2*y + cblock(relu(cblock(y))); relu except last) ----
    build_ysw<<<grid1(294912), blk, 0, stream>>>(yH1, bufA);
    const size_t msl = (size_t)4 * 256 * 72 * 72;  // one (4,256,72,72) slice
    for (int m = 0; m < 3; ++m) {
        const float* yin = (m % 2 == 0) ? bufA : bufB;
        float* yout      = (m % 2 == 0) ? bufB : bufA;
        cblock_wmma<<<dim3(5, 256), dim3(32), 0, stream>>>(
            yin, Mw + (size_t)(m * 2 + 0) * msl, tmpT, nullptr, 256, 72, 72, 0, 1);
        cblock_wmma<<<dim3(5, 256), dim3(32), 0, stream>>>(
            tmpT, Mw + (size_t)(m * 2 + 1) * msl, yout, yin, 256, 72, 72, 1, (m < 2) ? 1 : 0);
    }
    // final resnet output is in bufB (A->B, B->A, A->B)

    // ---- G stages (k=4 fold before G0 is a contiguous no-op) ----
    cblock_wmma<<<dim3(18, 64), dim3(32), 0, stream>>>(bufB, G0w, yG0, nullptr, 64, 288, 288, 0, 0);
    build_xG1<<<grid1(294912), blk, 0, stream>>>(yG0, xG1);
    cblock_wmma<<<dim3(12, 64), dim3(32), 0, stream>>>(xG1, G1w, yG1, nullptr, 64, 288, 192, 0, 0);

    // ---- U (perm(l=3)=identity and (16,64,192)->(16,256,48) is contiguous) ----
    cblock_wmma<<<dim3(4, 256), dim3(32), 0, stream>>>(yG1, Uw, yU, nullptr, 256, 48, 64, 0, 0);

    // ---- morton + CNN ----
    morton_gather<<<grid1(131072), blk, 0, stream>>>(yU, img0);
    conv2x2<<<grid1(8 * 128 * 128 * 6), blk, 0, stream>>>(img0, cw0, cb0, img1, 1, 6, 1);
    conv2x2<<<grid1(8 * 128 * 128 * 6), blk, 0, stream>>>(img1, cw1, cb1, img2, 6, 6, 1);
    conv2x2<<<grid1(8 * 128 * 128 * 1), blk, 0, stream>>>(img2, fw, fb, out, 6, 1, 0);
}